// GraphPool_61984968015931
// MI455X (gfx1250) — compile-verified
//
#include <hip/hip_runtime.h>
#include <math.h>

#define B_ 128
#define N_ 256
#define H_ 512
#define NH_ 8
#define K_ 5
#define LN_EPS 1e-5f

typedef __attribute__((ext_vector_type(2))) float v2f;
typedef __attribute__((ext_vector_type(8))) float v8f;

__global__ void init_zero_kernel(float* p, int n) {
    int i = blockIdx.x * blockDim.x + threadIdx.x;
    if (i < n) p[i] = 0.0f;
}

// scores[b,j] += sum_i w_i * attn[b,h,i,j]
// one block per (b,h), 512 threads = 16 waves, wave w handles columns [16w, 16w+16)
// D = A(16x4) x B(4x16) + C with A rows all equal to the weight chunk -> every D row
// is the partial column sum; accumulate over 64 i-chunks.
__global__ void scores_wmma_kernel(const float* __restrict__ attn,
                                   const float* __restrict__ x_mask,
                                   float* __restrict__ scores) {
    const int bh = blockIdx.x;          // b*NH + h
    const int b  = bh >> 3;
    __shared__ float s_w[N_];
    const int tid = threadIdx.x;
    if (tid < N_)
        s_w[tid] = (x_mask[b * N_ + tid] > 0.5f) ? (1.0f / (float)NH_) : 0.0f;
    __syncthreads();

    const int lane = tid & 31;
    const int wave = tid >> 5;              // 0..15
    const int jn   = (wave << 4) + (lane & 15);
    const int rsel = (lane >> 4) << 1;      // lanes 0-15 -> K=0,1 ; lanes 16-31 -> K=2,3
    const float* base = attn + (size_t)bh * N_ * N_;

    v8f c = {};
    #pragma unroll 8
    for (int ib = 0; ib < N_; ib += 4) {
        v2f a, bm;
        a.x  = s_w[ib + rsel];
        a.y  = s_w[ib + rsel + 1];
        bm.x = base[(size_t)(ib + rsel)     * N_ + jn];
        bm.y = base[(size_t)(ib + rsel + 1) * N_ + jn];
        c = __builtin_amdgcn_wmma_f32_16x16x4_f32(
                false, a, false, bm, (short)0, c, false, false);
    }
    // D row 0 (lanes 0-15, c[0]) holds the column sums; lanes 16-31 hold the
    // identical row M=8 -> only lanes 0-15 commit to avoid double counting.
    if (lane < 16) atomicAdd(&scores[b * N_ + jn], c[0]);
}

// dmax[b] = max over valid (i,j) of spatial_pos[b,i,j].
// Grid = B*8, each block handles a 32-row chunk; commit via uint atomicMax
// (valid: spatial_pos >= 0, so float order == uint-bit order; ws pre-zeroed).
__global__ void dmax_kernel(const float* __restrict__ sp,
                            const float* __restrict__ x_mask,
                            float* __restrict__ dmax_buf) {
    const int b     = blockIdx.x >> 3;
    const int chunk = blockIdx.x & 7;
    const int j     = threadIdx.x;          // 0..255
    __shared__ float s_valid[N_];
    __shared__ float s_red[N_];
    s_valid[j] = x_mask[b * N_ + j];
    __syncthreads();
    const bool valid = s_valid[j] > 0.5f;
    const float* spb = sp + (size_t)b * N_ * N_;

    float mx = 0.0f;                        // safe identity: all values >= 0
    const int i0 = chunk * 32;
    for (int i = i0; i < i0 + 32; ++i) {
        if (s_valid[i] > 0.5f) {            // uniform across block
            float v = spb[(size_t)i * N_ + j];
            if (valid) mx = fmaxf(mx, v);
        }
    }
    s_red[j] = mx;
    __syncthreads();
    for (int s = N_ / 2; s > 0; s >>= 1) {
        if (j < s) s_red[j] = fmaxf(s_red[j], s_red[j + s]);
        __syncthreads();
    }
    if (j == 0)
        atomicMax((unsigned int*)&dmax_buf[b], __float_as_uint(s_red[0]));
}

// Per-batch: m, smax, farthest-point sampling, emit weight vector.
__global__ void fps_kernel(const float* __restrict__ sp,
                           const float* __restrict__ scores,
                           const float* __restrict__ x_mask,
                           const float* __restrict__ dmax_buf,
                           float* __restrict__ w_out) {
    const int b = blockIdx.x;
    const int j = threadIdx.x;              // 0..255
    __shared__ float s_red[N_];
    __shared__ int   s_idx[N_];

    const float vm = x_mask[b * N_ + j];
    const bool valid = vm > 0.5f;
    const float dmax = dmax_buf[b];

    // m = number of valid nodes
    s_red[j] = valid ? 1.0f : 0.0f;
    __syncthreads();
    for (int s = N_ / 2; s > 0; s >>= 1) {
        if (j < s) s_red[j] += s_red[j + s];
        __syncthreads();
    }
    const int m = (int)(s_red[0] + 0.5f);
    __syncthreads();

    // smax = max over valid j of scores[b,j]
    const float sc = scores[b * N_ + j];
    s_red[j] = valid ? sc : -INFINITY;
    __syncthreads();
    for (int s = N_ / 2; s > 0; s >>= 1) {
        if (j < s) s_red[j] = fmaxf(s_red[j], s_red[j + s]);
        __syncthreads();
    }
    const float smax = s_red[0];
    __syncthreads();

    // D[b,i,j] = sp[b,i,j]/dmax + scores[b,i]/smax*0.1 ; min_dist over i
    const float* spb = sp + (size_t)b * N_ * N_;
    const float tterm = sc / smax * 0.1f;
    float md = spb[(size_t)j * N_] / dmax + tterm;   // D[b,j,0]
    bool sel = (j == 0);

    for (int it = 0; it < K_ - 1; ++it) {
        float cand = (valid && !sel) ? md : -INFINITY;
        s_red[j] = cand;
        s_idx[j] = j;
        __syncthreads();
        for (int s = N_ / 2; s > 0; s >>= 1) {
            if (j < s) {
                float v2 = s_red[j + s];
                int   i2 = s_idx[j + s];
                // jnp.argmax: first (lowest-index) maximum wins
                if (v2 > s_red[j] || (v2 == s_red[j] && i2 < s_idx[j])) {
                    s_red[j] = v2;
                    s_idx[j] = i2;
                }
            }
            __syncthreads();
        }
        const int nw = s_idx[0];
        __syncthreads();
        if (j == nw) sel = true;
        float dnew = spb[(size_t)j * N_ + nw] / dmax + tterm;
        md = fminf(md, dnew);
    }

    float w;
    if (m <= K_) w = valid ? (1.0f / (float)m) : 0.0f;
    else         w = sel   ? (1.0f / (float)K_) : 0.0f;
    w_out[b * N_ + j] = w;
}

// rep[b,h] = sum_n w[b,n]*x[b,n,h], then LayerNorm over H. One block per batch.
__global__ void rep_ln_kernel(const float* __restrict__ x,
                              const float* __restrict__ w,
                              const float* __restrict__ gamma,
                              const float* __restrict__ beta,
                              float* __restrict__ out) {
    const int b = blockIdx.x;
    const int h = threadIdx.x;              // 0..511
    __shared__ float s_w[N_];
    __shared__ float s_red[H_];
    if (h < N_) s_w[h] = w[b * N_ + h];
    __syncthreads();

    const float* xb = x + (size_t)b * N_ * H_;
    float acc = 0.0f;
    for (int n = 0; n < N_; ++n) {
        float wn = s_w[n];                  // uniform across block
        if (wn != 0.0f) acc = fmaf(wn, xb[(size_t)n * H_ + h], acc);
    }

    s_red[h] = acc;
    __syncthreads();
    for (int s = H_ / 2; s > 0; s >>= 1) {
        if (h < s) s_red[h] += s_red[h + s];
        __syncthreads();
    }
    const float mu = s_red[0] * (1.0f / (float)H_);
    __syncthreads();

    const float d = acc - mu;
    s_red[h] = d * d;
    __syncthreads();
    for (int s = H_ / 2; s > 0; s >>= 1) {
        if (h < s) s_red[h] += s_red[h + s];
        __syncthreads();
    }
    const float var = s_red[0] * (1.0f / (float)H_);
    const float inv = 1.0f / sqrtf(var + LN_EPS);
    out[b * H_ + h] = d * inv * gamma[h] + beta[h];
}

extern "C" void kernel_launch(void* const* d_in, const int* in_sizes, int n_in,
                              void* d_out, int out_size, void* d_ws, size_t ws_size,
                              hipStream_t stream) {
    const float* x     = (const float*)d_in[0];
    const float* attn  = (const float*)d_in[1];
    const float* sp    = (const float*)d_in[2];
    const float* xmask = (const float*)d_in[3];
    const float* gamma = (const float*)d_in[4];
    const float* beta  = (const float*)d_in[5];
    float* out = (float*)d_out;

    float* scores   = (float*)d_ws;             // B*N floats
    float* dmax_buf = scores + B_ * N_;         // B floats (contiguous for init)
    float* wbuf     = dmax_buf + B_;            // B*N floats

    const int n_init = B_ * N_ + B_;            // scores + dmax, contiguous
    init_zero_kernel<<<(n_init + 255) / 256, 256, 0, stream>>>(scores, n_init);
    dmax_kernel<<<B_ * 8, N_, 0, stream>>>(sp, xmask, dmax_buf);
    scores_wmma_kernel<<<B_ * NH_, 512, 0, stream>>>(attn, xmask, scores);
    fps_kernel<<<B_, N_, 0, stream>>>(sp, scores, xmask, dmax_buf, wbuf);
    rep_ln_kernel<<<B_, H_, 0, stream>>>(x, wbuf, gamma, beta, out);
}